// DynamicPartitionMaskStitchModule_11098195493301
// MI455X (gfx1250) — compile-verified
//
#include <hip/hip_runtime.h>
#include <hip/hip_bf16.h>
#include <stdint.h>

// The reference op (dynamic_partition -> dynamic_stitch over a full
// permutation) is mathematically the identity: out == data. This is a pure
// HBM-bandwidth copy: ~1.07 GB traffic, ~46 us floor at 23.3 TB/s.
// Main path: CDNA5 async global<->LDS DMA (ASYNCcnt) with b128 transactions,
// so data never round-trips through VGPRs.

#ifndef __has_builtin
#define __has_builtin(x) 0
#endif

#if __has_builtin(__builtin_amdgcn_global_load_async_to_lds_b128) && \
    __has_builtin(__builtin_amdgcn_global_store_async_from_lds_b128)
#define USE_ASYNC_LDS_COPY 1
#else
#define USE_ASYNC_LDS_COPY 0
#endif

// Native vector + address-space-qualified types for the builtins.
typedef float v4f __attribute__((ext_vector_type(4)));
typedef int v4i __attribute__((ext_vector_type(4)));
typedef __attribute__((address_space(1))) v4i gv4i;  // global int4
typedef __attribute__((address_space(3))) v4i sv4i;  // LDS int4

namespace {
constexpr int TPB = 256;           // 8 wave32 per block
constexpr int V4_PER_THREAD = 4;   // 64 bytes per thread per tile
constexpr int TILE_V4 = TPB * V4_PER_THREAD;  // 1024 x 16B = 16 KB / block
}  // namespace

__device__ __forceinline__ void wait_asynccnt0() {
#if __has_builtin(__builtin_amdgcn_s_wait_asynccnt)
  __builtin_amdgcn_s_wait_asynccnt(0);
#else
  asm volatile("s_wait_asynccnt 0" ::: "memory");
#endif
}

__global__ void __launch_bounds__(TPB)
identity_copy_kernel(const v4f* __restrict__ src, v4f* __restrict__ dst,
                     long long n4, const float* __restrict__ src_s,
                     float* __restrict__ dst_s, long long n) {
  const long long base = (long long)blockIdx.x * TILE_V4;
  const int tid = threadIdx.x;

#if USE_ASYNC_LDS_COPY
  __shared__ v4f stage[TILE_V4];  // 16 KB staging (320 KB/WGP available)
  if (base + TILE_V4 <= n4) {
    // DMA-style: global -> LDS (async b128), s_wait_asynccnt 0, LDS -> global.
    // Each lane owns its own 16 B slot, so the wave-level async counter is
    // the only synchronization needed (no workgroup barrier).
#pragma unroll
    for (int i = 0; i < V4_PER_THREAD; ++i) {
      const long long idx = base + (long long)i * TPB + tid;
      __builtin_amdgcn_global_load_async_to_lds_b128(
          (gv4i*)(uintptr_t)(src + idx),
          (sv4i*)(uint32_t)(uintptr_t)(&stage[i * TPB + tid]),
          /*offset=*/0, /*cpol=*/0);
    }
    wait_asynccnt0();
#pragma unroll
    for (int i = 0; i < V4_PER_THREAD; ++i) {
      const long long idx = base + (long long)i * TPB + tid;
      __builtin_amdgcn_global_store_async_from_lds_b128(
          (gv4i*)(uintptr_t)(dst + idx),
          (sv4i*)(uint32_t)(uintptr_t)(&stage[i * TPB + tid]),
          /*offset=*/0, /*cpol=*/0);
    }
    wait_asynccnt0();
  } else
#endif
  {
    // Boundary tile / fallback: b128 non-temporal copy through VGPRs.
    // (Data streams once and exceeds the 192 MB L2 -> NT hints.)
#pragma unroll
    for (int i = 0; i < V4_PER_THREAD; ++i) {
      const long long idx = base + (long long)i * TPB + tid;
      if (idx < n4) {
        v4f v = __builtin_nontemporal_load(src + idx);
        __builtin_nontemporal_store(v, dst + idx);
      }
    }
  }

  // Scalar tail for n % 4 elements (none for N*D = 128e6, but stay generic).
  if (blockIdx.x == 0 && tid == 0) {
    for (long long e = n4 * 4; e < n; ++e) dst_s[e] = src_s[e];
  }
}

extern "C" void kernel_launch(void* const* d_in, const int* in_sizes, int n_in,
                              void* d_out, int out_size, void* d_ws,
                              size_t ws_size, hipStream_t stream) {
  (void)n_in;
  (void)d_ws;
  (void)ws_size;
  (void)out_size;
  const float* data = (const float*)d_in[0];  // [N, D] fp32
  // d_in[1] (partitions) is provably irrelevant: the op is the identity.
  float* out = (float*)d_out;

  const long long n = (long long)in_sizes[0];  // N*D elements
  const long long n4 = n / 4;                  // float4 count
  long long blocks = (n4 + TILE_V4 - 1) / TILE_V4;
  if (blocks < 1) blocks = 1;  // still run the scalar-tail path

  identity_copy_kernel<<<(uint32_t)blocks, TPB, 0, stream>>>(
      (const v4f*)data, (v4f*)out, n4, data, out, n);
}